// SingleHeadAggregation_79001628443119
// MI455X (gfx1250) — compile-verified
//
#include <hip/hip_runtime.h>

typedef __attribute__((ext_vector_type(2))) float v2f;
typedef __attribute__((ext_vector_type(8))) float v8f;

// Problem constants from the reference
constexpr int Bn = 128;   // batch
constexpr int Gn = 1024;  // graph nodes
constexpr int Dn = 128;   // feature dim
constexpr int TILE_N = 16;
constexpr int WAVES = 4;             // waves per block
constexpr int KCHUNK = Gn / WAVES;   // 256 K per wave
constexpr int WSTRIDE = Gn + 21;     // LDS row stride, bank-skewed (21 % 64 != 0)

__global__ __launch_bounds__(32 * WAVES)
void SingleHeadAggregation_kernel(const float* __restrict__ h,
                                  const float* __restrict__ adj,
                                  const int*   __restrict__ phone,
                                  float* __restrict__ out) {
    // rows: [0] = row*maskBF, [1] = row*maskAF, [2] = zeros (for inactive A lanes)
    __shared__ float wlds[3 * WSTRIDE];
    __shared__ float red[WAVES][2][TILE_N];

    const int blk  = blockIdx.x;
    const int i    = blk >> 3;          // batch index  (Dn/TILE_N == 8 tiles)
    const int tile = blk & 7;
    const int d0   = tile * TILE_N;

    const int tid  = threadIdx.x;
    const int wave = tid >> 5;
    const int lane = tid & 31;
    const int n    = lane & 15;         // N (B-matrix) / M (A-matrix) sub-index
    const int kup  = (lane >> 4) << 1;  // K offset: lanes 16-31 carry K+2

    const int p = phone[i];             // pivot

    // ---- stage pre-masked weight rows into LDS (one-time; masks leave the loop) ----
    {
        float* wbf = wlds;
        float* waf = wlds + WSTRIDE;
        float* wz  = wlds + 2 * WSTRIDE;
        const float* rowp = adj + (size_t)i * Gn * Gn + (size_t)p * Gn;
        #pragma unroll
        for (int j = tid; j < Gn; j += 32 * WAVES) {
            const float r = rowp[j];            // coalesced b32
            wbf[j] = (j < p) ? r : 0.0f;
            waf[j] = (j > p) ? r : 0.0f;
        }
        #pragma unroll
        for (int j = tid; j < WSTRIDE; j += 32 * WAVES) wz[j] = 0.0f;
    }
    __syncthreads();

    // per-lane A source: row 0 -> wbf, row 1 -> waf, rows 2..15 -> zero row
    const int rsel = (n < 2) ? n : 2;
    const float* wl = wlds + rsel * WSTRIDE + kup;

    // per-lane B source column (kup folded into the base)
    const float* hb = h + (size_t)i * Gn * Dn + (size_t)kup * Dn + (size_t)d0 + (size_t)n;

    v8f c0 = {};
    v8f c1 = {};

    const int kstart = wave * KCHUNK;
    #pragma unroll 4
    for (int k0 = kstart; k0 < kstart + KCHUNK; k0 += 8) {
        // ---- K block [k0, k0+4) -> accumulator c0 ----
        {
            const float b0v = hb[(size_t)k0 * Dn];
            const float b1v = hb[(size_t)(k0 + 1) * Dn];
            v2f a = { wl[k0], wl[k0 + 1] };
            v2f b = { b0v, b1v };
            c0 = __builtin_amdgcn_wmma_f32_16x16x4_f32(false, a, false, b,
                                                       (short)0, c0, false, false);
        }
        // ---- K block [k0+4, k0+8) -> accumulator c1 (breaks RAW chain) ----
        {
            const float b0v = hb[(size_t)(k0 + 4) * Dn];
            const float b1v = hb[(size_t)(k0 + 5) * Dn];
            v2f a = { wl[k0 + 4], wl[k0 + 5] };
            v2f b = { b0v, b1v };
            c1 = __builtin_amdgcn_wmma_f32_16x16x4_f32(false, a, false, b,
                                                       (short)0, c1, false, false);
        }
    }

    // combine accumulators; only C rows M=0 (VGPR0) and M=1 (VGPR1) matter
    const float gbf = c0[0] + c1[0];   // M=0, N=lane (lanes 0-15)
    const float gaf = c0[1] + c1[1];   // M=1, N=lane (lanes 0-15)

    if (lane < 16) {
        red[wave][0][n] = gbf;
        red[wave][1][n] = gaf;
    }
    __syncthreads();

    // deterministic cross-wave reduction + store (one writer per output element)
    if (tid < 32) {
        const int sel = tid >> 4;      // 0 = g_bf, 1 = g_af
        const int nn  = tid & 15;
        const float s = red[0][sel][nn] + red[1][sel][nn] +
                        red[2][sel][nn] + red[3][sel][nn];
        out[(size_t)sel * Bn * Dn + (size_t)i * Dn + (size_t)d0 + (size_t)nn] = s;
    }
}

extern "C" void kernel_launch(void* const* d_in, const int* in_sizes, int n_in,
                              void* d_out, int out_size, void* d_ws, size_t ws_size,
                              hipStream_t stream) {
    (void)in_sizes; (void)n_in; (void)out_size; (void)d_ws; (void)ws_size;
    const float* h     = (const float*)d_in[0];
    const float* adj   = (const float*)d_in[1];
    const int*   phone = (const int*)d_in[2];
    float*       out   = (float*)d_out;

    dim3 grid(Bn * (Dn / TILE_N));   // 1024 blocks: (batch, 16-wide D tile)
    dim3 block(32 * WAVES);          // 4 waves
    SingleHeadAggregation_kernel<<<grid, block, 0, stream>>>(h, adj, phone, out);
}